// GraphAttentionLayer_43928925503993
// MI455X (gfx1250) — compile-verified
//
#include <hip/hip_runtime.h>
#include <math.h>

typedef __attribute__((ext_vector_type(2))) float v2f;
typedef __attribute__((ext_vector_type(8))) float v8f;

#define NEGC  (-9e15f)
#define SLOPE 0.2f

__device__ __forceinline__ float waveMax(float v) {
#pragma unroll
  for (int off = 16; off > 0; off >>= 1)
    v = fmaxf(v, __shfl_xor(v, off, 32));
  return v;
}
__device__ __forceinline__ float waveSum(float v) {
#pragma unroll
  for (int off = 16; off > 0; off >>= 1)
    v += __shfl_xor(v, off, 32);
  return v;
}

// One block = (batch b, 16-row tile i0..i0+15). 256 threads = 8 wave32.
// Phase 1: per-wave row softmax reductions over j (two rows per wave).
// Phase 2: rank-1 outputs via v_wmma_f32_16x16x4_f32 outer products + ELU.
__global__ __launch_bounds__(256) void gat_fused_kernel(
    const float* __restrict__ input, const int* __restrict__ adj,
    const float* __restrict__ ext,   const float* __restrict__ side,
    const float* __restrict__ W,     const float* __restrict__ a,
    const float* __restrict__ WS,    const float* __restrict__ aS,
    const float* __restrict__ WQ,    const float* __restrict__ WK,
    const float* __restrict__ WV,    float* __restrict__ out)
{
  constexpr int N = 512, FX = 8, F = 64;
  __shared__ float xs[N], ys[N], exts[N * FX];
  __shared__ float cst[8];
  __shared__ float zs[16], wvs[16];

  const int b    = blockIdx.x >> 5;          // 8 batches
  const int i0   = (blockIdx.x & 31) * 16;   // 32 row tiles
  const int tid  = threadIdx.x;
  const int lane = tid & 31;
  const int wave = tid >> 5;

  // ---- cooperative LDS fill -------------------------------------------
  for (int j = tid; j < N; j += 256) {
    xs[j] = input[b * N + j];
    ys[j] = side[b * N + j];
  }
  for (int t = tid; t < N * FX; t += 256) exts[t] = ext[b * N * FX + t];
  if (tid == 0) {
    float c1 = 0.f, c2 = 0.f, d1 = 0.f, d2 = 0.f;
    for (int f = 0; f < F; ++f) {
      c1 += W[f]  * a[f];    c2 += W[f]  * a[F + f];
      d1 += WS[f] * aS[f];   d2 += WS[f] * aS[F + f];
    }
    float A0 = 0.f, B0 = 0.f, A1 = 0.f, B1 = 0.f;
    for (int d = 0; d < 16; ++d) {
      A0 += WQ[d]      * WK[d];       B0 += WQ[16 + d] * WK[d];
      A1 += WQ[d]      * WK[16 + d];  B1 += WQ[16 + d] * WK[16 + d];
    }
    cst[0] = c1; cst[1] = c2; cst[2] = d1; cst[3] = d2;
    cst[4] = A0 * 0.25f; cst[5] = B0 * 0.25f;   // 1/sqrt(D), D=16
    cst[6] = A1 * 0.25f; cst[7] = B1 * 0.25f;
  }
  __syncthreads();

  const float c1 = cst[0], c2 = cst[1], d1 = cst[2], d2 = cst[3];
  const float A0 = cst[4], B0 = cst[5], A1 = cst[6], B1 = cst[7];

  // ---- Phase 1: two rows per wave -------------------------------------
  for (int rep = 0; rep < 2; ++rep) {
    const int il = wave + rep * 8;       // 0..15 within tile
    const int i  = i0 + il;
    const float xi = xs[i], yi = ys[i];
    const int* __restrict__ adjRow = adj + i * N;
    float ei[FX];
#pragma unroll
    for (int f = 0; f < FX; ++f) ei[f] = exts[i * FX + f];

    // pass 1: row max of masked attention scores
    float m = NEGC;
    for (int j = lane; j < N; j += 32) {
      float e1 = xi * c1 + xs[j] * c2; e1 = (e1 >= 0.f) ? e1 : SLOPE * e1;
      float e2 = yi * d1 + ys[j] * d2; e2 = (e2 >= 0.f) ? e2 : SLOPE * e2;
      float s  = e1 + e2;
      float att = (adjRow[j] > 0 && s > 0.f) ? s : NEGC;
      m = fmaxf(m, att);
    }
    m = waveMax(m);

    // pass 2: exp-sum + weighted accumulations
    float S = 0.f, zacc = 0.f, wacc = 0.f;
    const float qa = xi * A0, qb = xi * A1;
    for (int j = lane; j < N; j += 32) {
      float e1 = xi * c1 + xs[j] * c2; e1 = (e1 >= 0.f) ? e1 : SLOPE * e1;
      float e2 = yi * d1 + ys[j] * d2; e2 = (e2 >= 0.f) ? e2 : SLOPE * e2;
      float s  = e1 + e2;
      float att = (adjRow[j] > 0 && s > 0.f) ? s : NEGC;
      float E = __expf(att - m);
      S    += E;
      zacc += E * xs[j];

      // inner softmax over FX=8 ext features
      float qk0 = qa + xs[j] * B0;
      float qk1 = qb + xs[j] * B1;
      float t[FX];
      float mf = NEGC;
#pragma unroll
      for (int f = 0; f < FX; ++f) {
        float dd = ei[f] * qk0 + exts[j * FX + f] * qk1;
        t[f] = (dd > 0.f) ? dd : NEGC;
        mf = fmaxf(mf, t[f]);
      }
      float ps = 0.f, pe = 0.f;
#pragma unroll
      for (int f = 0; f < FX; ++f) {
        float p = __expf(t[f] - mf);
        ps += p;
        pe += ei[f] * p;
      }
      wacc += E * (pe / ps);
    }
    S    = waveSum(S);
    zacc = waveSum(zacc);
    wacc = waveSum(wacc);
    if (lane == 0) { zs[il] = zacc / S; wvs[il] = wacc / S; }
  }
  __syncthreads();

  // ---- Phase 2: rank-1 outputs via WMMA outer product + ELU -----------
  // 8 units: sel = wave>>2 chooses (h' with W) vs (ext_rep with WV),
  // chunk = wave&3 chooses 16-wide slice of F=64. All 32 lanes of every
  // wave are active -> EXEC all-1s for the WMMA.
  {
    const int sel   = wave >> 2;
    const int chunk = wave & 3;
    const float* mat = sel ? WV : W;
    float av = 0.f, bv = 0.f;
    if (lane < 16) {
      av = sel ? wvs[lane] : zs[lane];      // A[m,0] = vec[m], all other K = 0
      bv = mat[chunk * 16 + lane];          // B[0,n] = mat[n], all other K = 0
    }
    v2f A = { av, 0.f };
    v2f B = { bv, 0.f };
    v8f C = {};
    v8f D = __builtin_amdgcn_wmma_f32_16x16x4_f32(
        /*neg_a=*/false, A, /*neg_b=*/false, B,
        /*c_mod=*/(short)0, C, /*reuse_a=*/false, /*reuse_b=*/false);

    const int mbase = (lane >> 4) * 8;   // lanes 16-31 hold M = 8..15
    const int n     = lane & 15;
    const size_t obase = (size_t)sel * (8ull * 512ull * 64ull);
#pragma unroll
    for (int r = 0; r < 8; ++r) {
      float v = D[r];
      v = (v > 0.f) ? v : (__expf(v) - 1.f);   // ELU, alpha = 1
      out[obase + ((size_t)(b * N + i0 + mbase + r)) * 64 + chunk * 16 + n] = v;
    }
  }
}

extern "C" void kernel_launch(void* const* d_in, const int* in_sizes, int n_in,
                              void* d_out, int out_size, void* d_ws, size_t ws_size,
                              hipStream_t stream) {
  (void)in_sizes; (void)n_in; (void)out_size; (void)d_ws; (void)ws_size;
  const float* input = (const float*)d_in[0];
  const int*   adj   = (const int*)  d_in[1];
  const float* ext   = (const float*)d_in[2];
  const float* side  = (const float*)d_in[3];
  const float* W     = (const float*)d_in[4];
  const float* a     = (const float*)d_in[5];
  const float* WS    = (const float*)d_in[6];
  const float* aS    = (const float*)d_in[7];
  const float* WQ    = (const float*)d_in[8];
  const float* WK    = (const float*)d_in[9];
  const float* WV    = (const float*)d_in[10];
  float* out = (float*)d_out;

  dim3 grid(256);   // 8 batches * 32 tiles of 16 rows
  dim3 block(256);  // 8 wave32
  gat_fused_kernel<<<grid, block, 0, stream>>>(input, adj, ext, side,
                                               W, a, WS, aS, WQ, WK, WV, out);
}